// DGLGlobalGraphLayer_72877005078595
// MI455X (gfx1250) — compile-verified
//
#include <hip/hip_runtime.h>
#include <hip/hip_bf16.h>
#include <math.h>

#define N_SRC 100000
#define N_DST 10000
#define NEDGE 320000
#define DIMD  128
#define LN_EPS 1e-5f

typedef float v2f __attribute__((ext_vector_type(2)));
typedef float v8f __attribute__((ext_vector_type(8)));

__device__ __forceinline__ float wave_sum(float v) {
#pragma unroll
  for (int o = 16; o > 0; o >>= 1) v += __shfl_xor(v, o, 32);
  return v;
}
__device__ __forceinline__ float wave_max(float v) {
#pragma unroll
  for (int o = 16; o > 0; o >>= 1) v = fmaxf(v, __shfl_xor(v, o, 32));
  return v;
}

// C[m0:m0+16*MT, :] = A[m0:m0+16*MT, 0:128] @ B[128,128]  (row-major f32)
// Block: 8 waves; wave w owns N-tile w and MT stacked 16x16 M-tiles that
// SHARE each B fragment (halves load-issue per WMMA for MT=2).
// V_WMMA_F32_16X16X4_F32; A staged in LDS, stride 132 (conflict-free).
template <int MT>
__global__ __launch_bounds__(256) void gemm128_kernel(
    const float* __restrict__ A, const float* __restrict__ B,
    float* __restrict__ C) {
  __shared__ float ldsA[16 * MT * 132];
  const int t = threadIdx.x;
  const int m0 = blockIdx.x * (16 * MT);

  // cooperative coalesced load of (16*MT)x128 A tile
#pragma unroll
  for (int i = 0; i < 8 * MT; ++i) {
    int idx = t + i * 256;
    int r = idx >> 7, c = idx & 127;
    ldsA[r * 132 + c] = A[(size_t)(m0 + r) * DIMD + c];
  }
  __syncthreads();

  const int lane = t & 31;
  const int wave = t >> 5;
  const int half = lane >> 4;  // 0: lanes 0-15, 1: lanes 16-31
  const int l16 = lane & 15;
  const int n0 = wave * 16;

  v8f acc[MT];
#pragma unroll
  for (int mt = 0; mt < MT; ++mt) acc[mt] = (v8f){};

#pragma unroll 4
  for (int k0 = 0; k0 < 128; k0 += 4) {
    // B frag: comp g holds K = k0+2*half+g, N = n0+l16 (shared by all M-tiles)
    v2f b;
    b.x = B[(size_t)(k0 + 2 * half) * DIMD + n0 + l16];
    b.y = B[(size_t)(k0 + 2 * half + 1) * DIMD + n0 + l16];
#pragma unroll
    for (int mt = 0; mt < MT; ++mt) {
      // A frag: lane holds M = l16 + 16*mt, K = k0+2*half+{0,1}
      v2f a = *(const v2f*)&ldsA[(l16 + 16 * mt) * 132 + k0 + 2 * half];
      acc[mt] = __builtin_amdgcn_wmma_f32_16x16x4_f32(
          false, a, false, b, (short)0, acc[mt], false, false);
    }
  }
  // C frag: comp g -> row m0 + 16*mt + g + 8*half, col n0 + l16
#pragma unroll
  for (int mt = 0; mt < MT; ++mt)
#pragma unroll
    for (int g = 0; g < 8; ++g)
      C[(size_t)(m0 + 16 * mt + g + 8 * half) * DIMD + n0 + l16] = acc[mt][g];
}

// raw[e] = edge_weight[e] * dot(tanh(p_src[src[e]] + p_dst[dst[e]]), w2)
// one wave per edge; lane owns 4 features (float4 coalesced).
__global__ __launch_bounds__(256) void edge_score_kernel(
    const float* __restrict__ p_src, const float* __restrict__ p_dst,
    const float* __restrict__ w2, const float* __restrict__ ew,
    const int* __restrict__ src_idx, const int* __restrict__ dst_idx,
    float* __restrict__ raw) {
  const int e = blockIdx.x * 8 + (threadIdx.x >> 5);
  const int lane = threadIdx.x & 31;
  const int cb = lane * 4;
  const int s = src_idx[e];
  const int d = dst_idx[e];
  float4 a = *(const float4*)&p_src[(size_t)s * DIMD + cb];
  float4 b = *(const float4*)&p_dst[(size_t)d * DIMD + cb];
  float4 w = *(const float4*)&w2[cb];
  float p = tanhf(a.x + b.x) * w.x + tanhf(a.y + b.y) * w.y +
            tanhf(a.z + b.z) * w.z + tanhf(a.w + b.w) * w.w;
  p = wave_sum(p);
  if (lane == 0) raw[e] = p * ew[e];
}

__device__ __forceinline__ int lower_bound_i(const int* __restrict__ arr,
                                             int n, int v) {
  int lo = 0, hi = n;
  while (lo < hi) {
    int mid = (lo + hi) >> 1;
    if (arr[mid] < v) lo = mid + 1; else hi = mid;
  }
  return lo;
}

// one wave per dst node: segment softmax over its (contiguous, dst sorted)
// edge range + weighted aggregation of h_src rows (gathers hit the 192MB L2).
__global__ __launch_bounds__(256) void softmax_agg_kernel(
    const float* __restrict__ raw, const int* __restrict__ src_idx,
    const int* __restrict__ dst_idx, const float* __restrict__ h_src,
    float* __restrict__ attn, float* __restrict__ h_glob) {
  const int d = (blockIdx.x * blockDim.x + threadIdx.x) >> 5;  // dst id
  const int lane = threadIdx.x & 31;
  const int lo = lower_bound_i(dst_idx, NEDGE, d);
  const int hi = lower_bound_i(dst_idx, NEDGE, d + 1);

  float m = -INFINITY;
  for (int e = lo + lane; e < hi; e += 32) m = fmaxf(m, raw[e]);
  m = wave_max(m);

  float s = 0.f;
  for (int e = lo + lane; e < hi; e += 32) s += __expf(raw[e] - m);
  s = wave_sum(s);
  const float inv = (hi > lo) ? 1.f / s : 0.f;

  // dim-parallel aggregation, serial over edges (avg degree ~32)
  const int cb = lane * 4;
  float4 acc = {0.f, 0.f, 0.f, 0.f};
  for (int e = lo; e < hi; ++e) {
    float a = __expf(raw[e] - m) * inv;  // uniform across lanes
    if (lane == 0) attn[e] = a;
    float4 hv = *(const float4*)&h_src[(size_t)src_idx[e] * DIMD + cb];
    acc.x += a * hv.x; acc.y += a * hv.y;
    acc.z += a * hv.z; acc.w += a * hv.w;
  }
  *(float4*)&h_glob[(size_t)d * DIMD + cb] = acc;
}

// out = [h_dst | h_glob] @ Wo + bo ; x = h_dst + out ; y = LayerNorm(x)
// block: 16 rows, 8 waves (one 16x16 N-tile each, K=256), then LN epilogue.
__global__ __launch_bounds__(256) void outproj_ln_kernel(
    const float* __restrict__ h_dst, const float* __restrict__ h_glob,
    const float* __restrict__ Wo, const float* __restrict__ bo,
    const float* __restrict__ gamma, const float* __restrict__ beta,
    float* __restrict__ y) {
  __shared__ float ldsA[16 * 260];  // concat tile, stride 260 (conflict-free)
  __shared__ float ldsO[16 * 132];  // GEMM result tile
  const int t = threadIdx.x;
  const int m0 = blockIdx.x * 16;

  // stage [h_dst | h_glob] 16x256 tile
#pragma unroll
  for (int i = 0; i < 16; ++i) {
    int idx = t + i * 256;
    int r = idx >> 8, c = idx & 255;
    float v = (c < 128) ? h_dst[(size_t)(m0 + r) * DIMD + c]
                        : h_glob[(size_t)(m0 + r) * DIMD + (c - 128)];
    ldsA[r * 260 + c] = v;
  }
  __syncthreads();

  const int lane = t & 31;
  const int wave = t >> 5;
  const int half = lane >> 4;
  const int l16 = lane & 15;
  const int n0 = wave * 16;

  v8f acc = {};
#pragma unroll 4
  for (int k0 = 0; k0 < 256; k0 += 4) {
    v2f a = *(const v2f*)&ldsA[l16 * 260 + k0 + 2 * half];
    v2f b;
    b.x = Wo[(size_t)(k0 + 2 * half) * DIMD + n0 + l16];
    b.y = Wo[(size_t)(k0 + 2 * half + 1) * DIMD + n0 + l16];
    acc = __builtin_amdgcn_wmma_f32_16x16x4_f32(false, a, false, b, (short)0,
                                                acc, false, false);
  }
#pragma unroll
  for (int g = 0; g < 8; ++g)
    ldsO[(g + 8 * half) * 132 + n0 + l16] = acc[g];
  __syncthreads();

  // LayerNorm: wave w handles rows 2w and 2w+1; lane owns 4 columns
  const int cb = lane * 4;
#pragma unroll
  for (int rr = 0; rr < 2; ++rr) {
    const int r = wave * 2 + rr;
    float x0 = ldsA[r * 260 + cb + 0] + ldsO[r * 132 + cb + 0] + bo[cb + 0];
    float x1 = ldsA[r * 260 + cb + 1] + ldsO[r * 132 + cb + 1] + bo[cb + 1];
    float x2 = ldsA[r * 260 + cb + 2] + ldsO[r * 132 + cb + 2] + bo[cb + 2];
    float x3 = ldsA[r * 260 + cb + 3] + ldsO[r * 132 + cb + 3] + bo[cb + 3];
    float sum = wave_sum(x0 + x1 + x2 + x3);
    float sq = wave_sum(x0 * x0 + x1 * x1 + x2 * x2 + x3 * x3);
    float mu = sum * (1.f / 128.f);
    float var = sq * (1.f / 128.f) - mu * mu;
    float rstd = rsqrtf(var + LN_EPS);
    float4 o;
    o.x = (x0 - mu) * rstd * gamma[cb + 0] + beta[cb + 0];
    o.y = (x1 - mu) * rstd * gamma[cb + 1] + beta[cb + 1];
    o.z = (x2 - mu) * rstd * gamma[cb + 2] + beta[cb + 2];
    o.w = (x3 - mu) * rstd * gamma[cb + 3] + beta[cb + 3];
    *(float4*)&y[(size_t)(m0 + r) * DIMD + cb] = o;
  }
}

extern "C" void kernel_launch(void* const* d_in, const int* in_sizes, int n_in,
                              void* d_out, int out_size, void* d_ws,
                              size_t ws_size, hipStream_t stream) {
  const float* h_src = (const float*)d_in[0];
  const float* h_dst = (const float*)d_in[1];
  const float* s_emb = (const float*)d_in[2];
  const float* ew    = (const float*)d_in[3];
  const int*   src_i = (const int*)d_in[4];
  const int*   dst_i = (const int*)d_in[5];
  const float* W1    = (const float*)d_in[6];  // [256,128]
  const float* w2    = (const float*)d_in[7];  // [128]
  const float* Wo    = (const float*)d_in[8];  // [256,128]
  const float* bo    = (const float*)d_in[9];
  const float* gamma = (const float*)d_in[10];
  const float* beta  = (const float*)d_in[11];

  // workspace layout (f32): p_src | p_dst | raw | h_glob  (~62.7 MB)
  float* p_src  = (float*)d_ws;
  float* p_dst  = p_src + (size_t)N_SRC * DIMD;
  float* raw    = p_dst + (size_t)N_DST * DIMD;
  float* h_glob = raw + NEDGE;

  // output layout: y [N_DST,128] then attn [E,1]
  float* y    = (float*)d_out;
  float* attn = y + (size_t)N_DST * DIMD;

  // 1) p_src = h_src @ W1[:128]   (32-row blocks, B frags shared by 2 M-tiles)
  gemm128_kernel<2><<<N_SRC / 32, 256, 0, stream>>>(h_src, W1, p_src);
  // 2) p_dst = s_emb @ W1[128:]
  gemm128_kernel<1><<<N_DST / 16, 256, 0, stream>>>(s_emb, W1 + 128 * DIMD,
                                                    p_dst);
  // 3) per-edge raw attention score
  edge_score_kernel<<<NEDGE / 8, 256, 0, stream>>>(p_src, p_dst, w2, ew,
                                                   src_i, dst_i, raw);
  // 4) segment softmax + weighted aggregation (one wave per dst)
  softmax_agg_kernel<<<N_DST / 8, 256, 0, stream>>>(raw, src_i, dst_i, h_src,
                                                    attn, h_glob);
  // 5) output projection + residual + LayerNorm
  outproj_ln_kernel<<<N_DST / 16, 256, 0, stream>>>(h_dst, h_glob, Wo, bo,
                                                    gamma, beta, y);
}